// AdditionLinearTorch_77189152243805
// MI455X (gfx1250) — compile-verified
//
#include <hip/hip_runtime.h>
#include <cstdint>

// y[b,o] = -sum_k |x[b,k] - W[o,k]| + bias[o]
// B=1024, K=512, O=512, fp32. L2-resident; VALU-bound; WMMA is mathematically
// inapplicable (|x-w| is not bilinear). CDNA5 feature used: async global->LDS
// DMA (GLOBAL_LOAD_ASYNC_TO_LDS_B128, ASYNCcnt) for double-buffered tiling.

#define BATCH 1024
#define IN_F  512
#define OUT_F 512

#define BM 64           // batch rows per block
#define BN 32           // output cols per block
#define BK 32           // k-chunk
#define LDR 36          // padded LDS row stride in floats (144B, 16B-aligned)
#define XS_SZ (BM * LDR)            // 2304 floats per x buffer
#define WS_SZ (BN * LDR)            // 1152 floats per w buffer
#define NCHUNK (IN_F / BK)          // 16

// ---- CDNA5 async global->LDS (tracked by ASYNCcnt) ----------------------
__device__ __forceinline__ void async_ld_b128(uint32_t lds_byte_off,
                                              const void* sbase,
                                              uint32_t gl_byte_off) {
  // GVS mode: mem_addr = SADDR64 + VADDR32; per-lane LDS dest in VDST.
  asm volatile("global_load_async_to_lds_b128 %0, %1, %2"
               :: "v"(lds_byte_off), "v"(gl_byte_off), "s"(sbase)
               : "memory");
}

__device__ __forceinline__ void wait_async0() {
  asm volatile("s_wait_asynccnt 0" ::: "memory");
}

__global__ __launch_bounds__(256)
void l1dist_bias_kernel(const float* __restrict__ x,
                        const float* __restrict__ w,
                        const float* __restrict__ bias,
                        float* __restrict__ out) {
  __shared__ __align__(16) float smem[2 * XS_SZ + 2 * WS_SZ];
  // Low 32 bits of the generic (flat) LDS address == DS-relative byte offset.
  const uint32_t lds_base = (uint32_t)(uintptr_t)&smem[0];

  const int t     = threadIdx.x;          // 0..255
  const int tileM = blockIdx.x * BM;
  const int tileN = blockIdx.y * BN;

  // ---- async load assignment: b128 = 4 floats per lane -------------------
  // xs tile: BM x BK floats = 512 quads -> 2 loads/thread
  // ws tile: BN x BK floats = 256 quads -> 1 load/thread
  const int xm0 = t >> 3;                 // 0..31
  const int xm1 = xm0 + 32;               // 32..63
  const int xk  = (t & 7) * 4;            // k-quad start within chunk
  const int wn  = t >> 3;                 // 0..31
  const int wk  = (t & 7) * 4;

  auto issue = [&](int c, int buf) {
    const int k0 = c * BK;
    const uint32_t xsB = lds_base + (uint32_t)(buf * XS_SZ) * 4u;
    const uint32_t wsB = lds_base + (uint32_t)(2 * XS_SZ + buf * WS_SZ) * 4u;
    async_ld_b128(xsB + (uint32_t)(xm0 * LDR + xk) * 4u, x,
                  (uint32_t)(((tileM + xm0) * IN_F) + k0 + xk) * 4u);
    async_ld_b128(xsB + (uint32_t)(xm1 * LDR + xk) * 4u, x,
                  (uint32_t)(((tileM + xm1) * IN_F) + k0 + xk) * 4u);
    async_ld_b128(wsB + (uint32_t)(wn * LDR + wk) * 4u, w,
                  (uint32_t)(((tileN + wn) * IN_F) + k0 + wk) * 4u);
  };

  // ---- compute mapping: 4(m) x 2(n) micro-tile per thread ----------------
  const int tm = t >> 4;                  // 0..15
  const int tn = t & 15;                  // 0..15
  const int m0 = tm * 4;
  const int n0 = tn * 2;

  float acc[4][2] = {};

  issue(0, 0);
  wait_async0();
  __syncthreads();

  for (int c = 0; c < NCHUNK; ++c) {
    const int buf = c & 1;
    if (c + 1 < NCHUNK) issue(c + 1, buf ^ 1);

    const float* xs = &smem[buf * XS_SZ];
    const float* ws = &smem[2 * XS_SZ + buf * WS_SZ];

#pragma unroll
    for (int kk = 0; kk < BK; kk += 4) {
      float4 a[4];
      float4 b[2];
#pragma unroll
      for (int j = 0; j < 4; ++j)
        a[j] = *reinterpret_cast<const float4*>(&xs[(m0 + j) * LDR + kk]);
#pragma unroll
      for (int i = 0; i < 2; ++i)
        b[i] = *reinterpret_cast<const float4*>(&ws[(n0 + i) * LDR + kk]);
#pragma unroll
      for (int j = 0; j < 4; ++j) {
#pragma unroll
        for (int i = 0; i < 2; ++i) {
          // v_sub_f32 + v_add_f32 with |src| modifier (abs is free in VOP3)
          acc[j][i] += __builtin_fabsf(a[j].x - b[i].x);
          acc[j][i] += __builtin_fabsf(a[j].y - b[i].y);
          acc[j][i] += __builtin_fabsf(a[j].z - b[i].z);
          acc[j][i] += __builtin_fabsf(a[j].w - b[i].w);
        }
      }
    }

    wait_async0();     // my wave's prefetch for chunk c+1 has landed
    __syncthreads();   // everyone done reading buf & writing buf^1
  }

  // ---- epilogue: out = bias - dist ---------------------------------------
  const float b0 = bias[tileN + n0 + 0];
  const float b1 = bias[tileN + n0 + 1];
#pragma unroll
  for (int j = 0; j < 4; ++j) {
    float2 r;
    r.x = b0 - acc[j][0];
    r.y = b1 - acc[j][1];
    *reinterpret_cast<float2*>(
        &out[(tileM + m0 + j) * OUT_F + tileN + n0]) = r;
  }
}

extern "C" void kernel_launch(void* const* d_in, const int* in_sizes, int n_in,
                              void* d_out, int out_size, void* d_ws, size_t ws_size,
                              hipStream_t stream) {
  (void)in_sizes; (void)n_in; (void)out_size; (void)d_ws; (void)ws_size;
  const float* x    = (const float*)d_in[0];
  const float* wgt  = (const float*)d_in[1];
  const float* bias = (const float*)d_in[2];
  float* out        = (float*)d_out;

  dim3 grid(BATCH / BM, OUT_F / BN);   // 16 x 16 = 256 blocks
  dim3 block(256);                     // 8 wave32
  l1dist_bias_kernel<<<grid, block, 0, stream>>>(x, wgt, bias, out);
}